// GCNNet_5978594476679
// MI455X (gfx1250) — compile-verified
//
#include <hip/hip_runtime.h>

#define N_NODES  100000
#define N_EDGES  1000000
#define DFEAT    64
#define N_GRAPHS 512

typedef __attribute__((ext_vector_type(2))) float v2f;
typedef __attribute__((ext_vector_type(8))) float v8f;

// ---------------------------------------------------------------------------
// Zero the per-graph accumulators (sum + count). Must run every launch since
// the harness does not re-poison between replays.
// ---------------------------------------------------------------------------
__global__ void gcn_zero_acc(float* __restrict__ out_sum, float* __restrict__ cnt) {
    int i = blockIdx.x * blockDim.x + threadIdx.x;
    if (i < N_GRAPHS) { out_sum[i] = 0.0f; cnt[i] = 0.0f; }
}

// ---------------------------------------------------------------------------
// s[v] = dot(x[v,:], W[:])  via V_WMMA_F32_16X16X4_F32.
// One wave handles a tile of 16 nodes. A = 16x4 slice of x
//   (ISA layout: lanes 0-15 -> M=lane, VGPR0/1 -> K=0,1; lanes 16-31 -> K=2,3)
// B = W broadcast into all 16 columns, so D[m][n] == s[m] for every n.
// After 16 accumulating WMMAs (K = 64), lane 0 holds s[0..7] in c[0..7] and
// lane 16 holds s[8..15].
// ---------------------------------------------------------------------------
__global__ void gcn_node_dot_wmma(const float* __restrict__ x,
                                  const float* __restrict__ W,
                                  float* __restrict__ s) {
    const int lane = threadIdx.x & 31;
    const int wave = threadIdx.x >> 5;
    const int tile = blockIdx.x * (blockDim.x >> 5) + wave;   // 16 nodes / tile
    if (tile >= (N_NODES / 16)) return;                        // wave-uniform

    const int row = lane & 15;          // M index for A
    const int kp  = (lane >> 4) << 1;   // K sub-offset: 0 (lanes 0-15) or 2
    const float* __restrict__ xrow = x + (size_t)(tile * 16 + row) * DFEAT;

    v8f c = {};
#pragma unroll
    for (int k = 0; k < DFEAT; k += 4) {
        v2f a, b;
        a.x = xrow[k + kp];
        a.y = xrow[k + kp + 1];
        b.x = W[k + kp];                // replicate W across all N columns
        b.y = W[k + kp + 1];
        // 8 args: (neg_a, A, neg_b, B, c_mod, C, reuse_a, reuse_b)
        c = __builtin_amdgcn_wmma_f32_16x16x4_f32(
                false, a, false, b, (short)0, c, false, false);
    }

    if (lane == 0 || lane == 16) {
        float* __restrict__ outp = s + tile * 16 + ((lane >> 4) << 3);
#pragma unroll
        for (int j = 0; j < 8; ++j) outp[j] = c[j];
    }
}

// ---------------------------------------------------------------------------
// counts[g] = #nodes with batch id g.  LDS histogram, then atomic flush.
// ---------------------------------------------------------------------------
__global__ void gcn_count_nodes(const long long* __restrict__ batch,
                                float* __restrict__ cnt) {
    __shared__ float lcnt[N_GRAPHS];
    for (int i = threadIdx.x; i < N_GRAPHS; i += blockDim.x) lcnt[i] = 0.0f;
    __syncthreads();

    const int stride = gridDim.x * blockDim.x;
    for (int i = blockIdx.x * blockDim.x + threadIdx.x; i < N_NODES; i += stride)
        atomicAdd(&lcnt[(int)batch[i]], 1.0f);

    __syncthreads();
    for (int i = threadIdx.x; i < N_GRAPHS; i += blockDim.x)
        if (lcnt[i] != 0.0f) atomicAdd(&cnt[i], lcnt[i]);
}

// ---------------------------------------------------------------------------
// Single pass over the 1M edges:
//   out_sum[batch[dst[e]]] += edge_attr[e] * s[src[e]]
// s (400 KB) and batch (800 KB) are L2-resident gathers; the per-graph
// accumulator lives in LDS (512 floats) and is flushed once per block.
// ---------------------------------------------------------------------------
__global__ void gcn_edge_accum(const long long* __restrict__ ei,
                               const float* __restrict__ ea,
                               const long long* __restrict__ batch,
                               const float* __restrict__ s,
                               float* __restrict__ out_sum) {
    __shared__ float lsum[N_GRAPHS];
    for (int i = threadIdx.x; i < N_GRAPHS; i += blockDim.x) lsum[i] = 0.0f;
    __syncthreads();

    const long long* __restrict__ src = ei;
    const long long* __restrict__ dst = ei + N_EDGES;
    const int stride = gridDim.x * blockDim.x;
    for (int e = blockIdx.x * blockDim.x + threadIdx.x; e < N_EDGES; e += stride) {
        const int u = (int)src[e];
        const int v = (int)dst[e];
        const float val = ea[e] * s[u];
        atomicAdd(&lsum[(int)batch[v]], val);
    }

    __syncthreads();
    for (int i = threadIdx.x; i < N_GRAPHS; i += blockDim.x)
        if (lsum[i] != 0.0f) atomicAdd(&out_sum[i], lsum[i]);
}

// ---------------------------------------------------------------------------
// out[g] = out_sum[g] / max(count[g], 1) + b
// ---------------------------------------------------------------------------
__global__ void gcn_finalize(const float* __restrict__ out_sum,
                             const float* __restrict__ cnt,
                             const float* __restrict__ bias,
                             float* __restrict__ out) {
    int g = blockIdx.x * blockDim.x + threadIdx.x;
    if (g < N_GRAPHS) out[g] = out_sum[g] / fmaxf(cnt[g], 1.0f) + bias[0];
}

extern "C" void kernel_launch(void* const* d_in, const int* in_sizes, int n_in,
                              void* d_out, int out_size, void* d_ws, size_t ws_size,
                              hipStream_t stream) {
    (void)in_sizes; (void)n_in; (void)out_size; (void)ws_size;

    const float*     x     = (const float*)d_in[0];       // [N_NODES, 64] f32
    const long long* ei    = (const long long*)d_in[1];   // [2, N_EDGES] i64
    const float*     ea    = (const float*)d_in[2];       // [N_EDGES] f32
    const long long* batch = (const long long*)d_in[3];   // [N_NODES] i64
    const float*     W     = (const float*)d_in[4];       // [1, 64] f32
    const float*     bias  = (const float*)d_in[5];       // [1] f32

    float* s       = (float*)d_ws;          // N_NODES floats (400 KB)
    float* out_sum = s + N_NODES;           // N_GRAPHS floats
    float* cnt     = out_sum + N_GRAPHS;    // N_GRAPHS floats
    float* out     = (float*)d_out;         // [N_GRAPHS, 1] f32

    gcn_zero_acc<<<(N_GRAPHS + 255) / 256, 256, 0, stream>>>(out_sum, cnt);

    // 6250 tiles of 16 nodes, 8 waves (256 threads) per block
    gcn_node_dot_wmma<<<(N_NODES / 16 + 7) / 8, 256, 0, stream>>>(x, W, s);

    gcn_count_nodes<<<256, 256, 0, stream>>>(batch, cnt);

    gcn_edge_accum<<<512, 256, 0, stream>>>(ei, ea, batch, s, out_sum);

    gcn_finalize<<<(N_GRAPHS + 255) / 256, 256, 0, stream>>>(out_sum, cnt, bias, out);
}